// rel_infer_train_27144193310750
// MI455X (gfx1250) — compile-verified
//
#include <hip/hip_runtime.h>
#include <hip/hip_bf16.h>

// MI455X / gfx1250: wave32, WMMA f32 16x16x4 (full-precision matrix path).
// Each wave computes a 2x2 block of 16x16 output tiles -> 2 loads per WMMA.
typedef float v2f __attribute__((ext_vector_type(2)));
typedef float v8f __attribute__((ext_vector_type(8)));

#define P_    256
#define C_    151
#define R_    51
#define CP_   160     // C padded to 10 tiles of 16
#define NPCH  4       // K(p)-dimension split for occupancy
#define PCHSZ (P_ / NPCH)

#define AOFF_Q1  (16 * R_)        // +16 q rows in rel[n,p,q,:]  (816 elem = 3264 B, fits IOFFSET)
#define A2OFF_Q1 (16 * P_ * R_)   // +16 q rows in rel[n,q,p,:]  (835584 B, fits 24-bit IOFFSET)

// thetaP layout: [pch][n*P+q][CP_]
__global__ void theta_gemm_kernel(const float* __restrict__ rel,     // [B,P,P,R]
                                  const float* __restrict__ rmat,    // [C,C,R]
                                  const int*   __restrict__ labels,  // [B*P]
                                  float* __restrict__ thetaP,
                                  int nrows)                         // B*P
{
    const int wv   = threadIdx.x >> 5;
    const int lane = threadIdx.x & 31;
    const int tile = blockIdx.x * (blockDim.x >> 5) + wv;  // ((n*8+qp)*5+cp)*4+pch
    const int pch  = tile & 3;
    const int cp   = (tile >> 2) % 5;          // c-tile pair: columns cp*32 .. +32
    const int qp   = (tile / 20) % 8;          // q-tile pair: rows    qp*32 .. +32
    const int n    = tile / 160;

    const int mrow = lane & 15;        // M (or N) index within a tile
    const int hi   = lane >> 4;        // lane half selects K placement
    const int q    = qp * 32 + mrow;   // q-tile 0 row; q-tile 1 = q+16 via imm offsets
    const int c0   = cp * 32 + mrow;
    const int c1   = c0 + 16;
    // Padded columns (c >= 151) compute a redundant copy of column 150; their
    // outputs land in the [151,160) pad and are never read downstream.
    const int ccl0 = (c0 < C_) ? c0 : (C_ - 1);
    const int ccl1 = (c1 < C_) ? c1 : (C_ - 1);

    const int p0 = pch * PCHSZ;
    const float* relbase = rel + (size_t)n * ((size_t)P_ * P_ * R_);
    const float* a1p  = relbase + ((size_t)p0 * P_ + q) * R_;   // rel[n,p,q,:], +P*R per p
    const float* a2p  = relbase + ((size_t)q * P_ + p0) * R_;   // rel[n,q,p,:], +R   per p
    const float* b1c0 = rmat + (size_t)ccl0 * R_;               // + lab*C*R per p
    const float* b1c1 = rmat + (size_t)ccl1 * R_;
    const float* b2c0 = rmat + (size_t)ccl0 * C_ * R_;          // + lab*R   per p
    const float* b2c1 = rmat + (size_t)ccl1 * C_ * R_;
    const int*   labp = labels + n * P_ + p0;

    // Per-lane constant weights (0.5*w_r folded into A; r=51 pad masked to 0).
    const int   rA0  = hi << 1;                 // A K-offset: 0 or 2
    const int   rB0  = hi;                      // B K-offset: 0 or 1
    const float w0   = hi ? 0.25f : 0.5f;       // a.x weight at r==0
    const float wAyL = hi ? 0.f   : 0.25f;      // last-step a.y (r=51 -> 0)
    const float wByL = hi ? 0.f   : 1.f;        // last-step b.y (r=51 -> 0)

    v8f accs[2][2];
#pragma unroll
    for (int qi = 0; qi < 2; ++qi)
#pragma unroll
        for (int ci = 0; ci < 2; ++ci)
            accs[qi][ci] = (v8f){0.f, 0.f, 0.f, 0.f, 0.f, 0.f, 0.f, 0.f};

    for (int p = 0; p < PCHSZ; ++p) {
        const int lab = __builtin_amdgcn_readfirstlane(labp[p]);   // wave-uniform
        const float* b1p0 = b1c0 + (size_t)lab * (C_ * R_);        // M[lab, c,  :]
        const float* b1p1 = b1c1 + (size_t)lab * (C_ * R_);
        const float* b2p0 = b2c0 + (size_t)lab * R_;               // M[c,  lab, :]
        const float* b2p1 = b2c1 + (size_t)lab * R_;

#define WMMA4(A, B0, B1, qi)                                                          \
        accs[qi][0] = __builtin_amdgcn_wmma_f32_16x16x4_f32(false, A, false, B0,      \
                                            (short)0, accs[qi][0], false, false);     \
        accs[qi][1] = __builtin_amdgcn_wmma_f32_16x16x4_f32(false, A, false, B1,      \
                                            (short)0, accs[qi][1], false, false);

#define WSTEP(iax, iay, wax, way, ibx, iby, wby)                                      \
        {                                                                             \
            v2f a10, a11, a20, a21, b10, b11, b20, b21;                               \
            a10.x = a1p[iax] * (wax);             a10.y = a1p[iay] * (way);           \
            a11.x = a1p[(iax) + AOFF_Q1] * (wax); a11.y = a1p[(iay) + AOFF_Q1] * (way);\
            a20.x = a2p[iax] * (wax);             a20.y = a2p[iay] * (way);           \
            a21.x = a2p[(iax) + A2OFF_Q1] * (wax);a21.y = a2p[(iay) + A2OFF_Q1] * (way);\
            b10.x = b1p0[ibx];  b10.y = b1p0[iby] * (wby);                            \
            b11.x = b1p1[ibx];  b11.y = b1p1[iby] * (wby);                            \
            b20.x = b2p0[ibx];  b20.y = b2p0[iby] * (wby);                            \
            b21.x = b2p1[ibx];  b21.y = b2p1[iby] * (wby);                            \
            WMMA4(a10, b10, b11, 0)                                                   \
            WMMA4(a20, b20, b21, 0)                                                   \
            WMMA4(a11, b10, b11, 1)                                                   \
            WMMA4(a21, b20, b21, 1)                                                   \
        }

        WSTEP(rA0, rA0 + 1, w0, 0.25f, rB0, rB0 + 2, 1.f)          // r0 = 0
#pragma unroll
        for (int r0 = 4; r0 < 48; r0 += 4)                          // r0 = 4..44
            WSTEP(r0 + rA0, r0 + rA0 + 1, 0.25f, 0.25f, r0 + rB0, r0 + rB0 + 2, 1.f)
        WSTEP(48 + rA0, 49 + hi, 0.25f, wAyL, 48 + rB0, 50, wByL)   // r0 = 48 (clamped)
#undef WSTEP
#undef WMMA4

        a1p += (size_t)P_ * R_;
        a2p += R_;
    }

    // Subtract diagonal p==q contribution (owned by the chunk containing q) and store.
    const bool ownsDiag = (pch == (qp >> 1));   // [qp*32, qp*32+32) lies in one 64-chunk
    const int* labv = labels + n * P_;
#pragma unroll
    for (int qi = 0; qi < 2; ++qi) {
#pragma unroll
        for (int v = 0; v < 8; ++v) {
            const int qq = qp * 32 + qi * 16 + v + hi * 8;  // D layout: M = v (+8 upper half)
#pragma unroll
            for (int ci = 0; ci < 2; ++ci) {
                const int cc = cp * 32 + ci * 16 + mrow;    // N = lane % 16
                float a = accs[qi][ci][v];
                if (ownsDiag && cc < C_) {
                    const int labq = labv[qq];
                    const float* drel = relbase + ((size_t)qq * P_ + qq) * R_;
                    const float* m1 = rmat + ((size_t)labq * C_ + cc) * R_;
                    const float* m2 = rmat + ((size_t)cc * C_ + labq) * R_;
                    float corr = drel[0] * 0.5f * (m1[0] + m2[0]);
                    for (int r = 1; r < R_; ++r)
                        corr += drel[r] * 0.25f * (m1[r] + m2[r]);
                    a -= corr;
                }
                thetaP[((size_t)pch * nrows + n * P_ + qq) * CP_ + cc] = a;
            }
        }
    }
}

// One wave per row: log-softmax over 151 classes + NLL at label (deterministic shuffles).
__global__ void nll_kernel(const float* __restrict__ thetaP,
                           const int* __restrict__ labels,
                           float* __restrict__ rowloss, int nrows)
{
    const int lane = threadIdx.x & 31;
    const int row  = blockIdx.x * (blockDim.x >> 5) + (threadIdx.x >> 5);
    if (row >= nrows) return;

    float tv[5];
#pragma unroll
    for (int k = 0; k < 5; ++k) {
        const int cc = lane + k * 32;
        float s = -1e30f;
        if (cc < C_) {
            s = 0.f;
            for (int ch = 0; ch < NPCH; ++ch)
                s += thetaP[((size_t)ch * nrows + row) * CP_ + cc];
        }
        tv[k] = s;
    }
    float mx = tv[0];
#pragma unroll
    for (int k = 1; k < 5; ++k) mx = fmaxf(mx, tv[k]);
    for (int off = 16; off; off >>= 1) mx = fmaxf(mx, __shfl_xor(mx, off, 32));

    float se = 0.f;
#pragma unroll
    for (int k = 0; k < 5; ++k) se += __expf(tv[k] - mx);  // exp(-1e30-mx)==0 pads
    for (int off = 16; off; off >>= 1) se += __shfl_xor(se, off, 32);

    if (lane == 0) {
        const int lab = labels[row];
        float tl = 0.f;
        for (int ch = 0; ch < NPCH; ++ch)
            tl += thetaP[((size_t)ch * nrows + row) * CP_ + lab];
        rowloss[row] = -(tl - mx - __logf(se));
    }
}

// Fixed-order single-block reduction -> mean (bitwise deterministic across replays).
__global__ void mean_kernel(const float* __restrict__ rowloss,
                            float* __restrict__ out, int nrows)
{
    __shared__ float sm[256];
    float s = 0.f;
    for (int i = threadIdx.x; i < nrows; i += 256) s += rowloss[i];
    sm[threadIdx.x] = s;
    __syncthreads();
    for (int stride = 128; stride; stride >>= 1) {
        if ((int)threadIdx.x < stride) sm[threadIdx.x] += sm[threadIdx.x + stride];
        __syncthreads();
    }
    if (threadIdx.x == 0) out[0] = sm[0] / (float)nrows;
}

extern "C" void kernel_launch(void* const* d_in, const int* in_sizes, int n_in,
                              void* d_out, int out_size, void* d_ws, size_t ws_size,
                              hipStream_t stream) {
    // d_in[0] = roi_scores (UNUSED by the reference result: _unary is dead code)
    const float* rel    = (const float*)d_in[1];   // [B*P*P, R]
    const float* rmat   = (const float*)d_in[2];   // [C, C, R]
    const int*   labels = (const int*)  d_in[3];   // [B*P]
    const int nrows = in_sizes[3];                 // B*P
    const int B     = nrows / P_;

    float* thetaP  = (float*)d_ws;                          // NPCH * nrows * CP_ floats
    float* rowloss = thetaP + (size_t)NPCH * nrows * CP_;   // nrows floats

    const int tiles = B * 8 * 5 * NPCH;   // waves: (q-pairs) x (c-pairs) x (p-chunks)
    theta_gemm_kernel<<<tiles / 4, 128, 0, stream>>>(rel, rmat, labels, thetaP, nrows);
    nll_kernel<<<(nrows + 3) / 4, 128, 0, stream>>>(thetaP, labels, rowloss, nrows);
    mean_kernel<<<1, 256, 0, stream>>>(rowloss, (float*)d_out, nrows);
}